// NetVLAD_45174466019667
// MI455X (gfx1250) — compile-verified
//
#include <hip/hip_runtime.h>
#include <cstdint>

typedef __attribute__((ext_vector_type(2))) float v2f;
typedef __attribute__((ext_vector_type(8))) float v8f;

#define NB 32
#define ND 128
#define NK 64
#define IJ 1024
#define ROWS 128
#define CHUNKS (IJ / ROWS)      // 8
#define DPAD (ND + 1)           // 129, avoid LDS bank conflicts
#define KPAD (NK + 1)           // 65

#define S1_ELEMS (NB * NK * ND) // 262144
#define S0_ELEMS (NB * NK)      // 2048
#define WS_ZERO_ELEMS (S1_ELEMS + S0_ELEMS + 1)

// Async global->LDS 16-byte copy (CDNA5 GLOBAL_LOAD_ASYNC_TO_LDS_B128,
// tracked by ASYNCcnt). Low 32 bits of a generic pointer to __shared__
// are the DS-relative LDS offset (ISA 10.2 aperture rules).
__device__ __forceinline__ void async_copy_b128(const void* gptr, void* lds) {
    uint32_t lds_off = (uint32_t)(uintptr_t)lds;
    asm volatile("global_load_async_to_lds_b128 %0, %1, off"
                 :: "v"(lds_off), "v"(gptr) : "memory");
}
__device__ __forceinline__ void wait_async_all() {
    asm volatile("s_wait_asynccnt 0x0" ::: "memory");
}

__global__ void netvlad_zero_ws(float* __restrict__ ws, int n) {
    int i = blockIdx.x * blockDim.x + threadIdx.x;
    if (i < n) ws[i] = 0.0f;
}

// One workgroup = (batch b, chunk of 128 ij rows).
// Computes sim = softmax(-||f-c||^2) for its rows entirely in LDS, then
// accumulates S1[b,k,d] += sim^T @ F and S0[b,k] += sum_rows sim.
__global__ __launch_bounds__(256) void netvlad_sim_acc(
    const float* __restrict__ feat,   // (B, IJ, D)
    const float* __restrict__ cent,   // (B, K, D)
    float* __restrict__ S1,           // (B, K, D) accumulators
    float* __restrict__ S0)           // (B, K)    accumulators
{
    __shared__ float sF[ROWS][DPAD];    // 128 x 129 f32
    __shared__ float sC[NK][DPAD];      // 64  x 129 f32
    __shared__ float sSim[ROWS][KPAD];  // 128 x 65  f32
    __shared__ float sF2[ROWS];
    __shared__ float sC2[NK];

    const int tid   = threadIdx.x;
    const int b     = blockIdx.x / CHUNKS;
    const int chunk = blockIdx.x % CHUNKS;
    const int row0  = chunk * ROWS;

    // ---- stage F chunk and C into LDS via async global->LDS b128 ----
    // Every 16B transfer stays inside one padded row (ND % 4 == 0).
    const float4* fsrc = (const float4*)(feat + ((size_t)b * IJ + row0) * ND);
    #pragma unroll
    for (int i = 0; i < (ROWS * ND / 4) / 256; ++i) {   // 16 iters
        int e4 = tid + i * 256;
        int e = e4 * 4;
        int r = e / ND, d = e % ND;
        async_copy_b128((const void*)(fsrc + e4), (void*)&sF[r][d]);
    }
    const float4* csrc = (const float4*)(cent + (size_t)b * NK * ND);
    #pragma unroll
    for (int i = 0; i < (NK * ND / 4) / 256; ++i) {     // 8 iters
        int e4 = tid + i * 256;
        int e = e4 * 4;
        int r = e / ND, d = e % ND;
        async_copy_b128((const void*)(csrc + e4), (void*)&sC[r][d]);
    }
    wait_async_all();     // each wave drains its own ASYNCcnt
    __syncthreads();      // then all waves agree LDS is filled

    // ---- row norms ||f||^2 and ||c||^2 ----
    if (tid < ROWS) {
        float s = 0.0f;
        for (int d = 0; d < ND; ++d) { float x = sF[tid][d]; s += x * x; }
        sF2[tid] = s;
    } else if (tid < ROWS + NK) {
        int k = tid - ROWS;
        float s = 0.0f;
        for (int d = 0; d < ND; ++d) { float x = sC[k][d]; s += x * x; }
        sC2[k] = s;
    }
    __syncthreads();

    const int wave  = tid >> 5;
    const int lane  = tid & 31;
    const int lhalf = lane >> 4;   // 0 or 1
    const int l15   = lane & 15;

    // ---- GEMM 1: fc[m][n] = sum_d F[m][d] * C[n][d] via V_WMMA_F32_16X16X4_F32
    //      wave w owns M-tile w (rows w*16..w*16+15), all 4 N-tiles.
    {
        v8f acc0 = {}, acc1 = {}, acc2 = {}, acc3 = {};
        const int m = wave * 16 + l15;
        for (int k0 = 0; k0 < ND; k0 += 4) {
            const int kb = k0 + (lhalf << 1);
            v2f a; a.x = sF[m][kb]; a.y = sF[m][kb + 1];
            v2f b0; b0.x = sC[0 * 16 + l15][kb]; b0.y = sC[0 * 16 + l15][kb + 1];
            v2f b1; b1.x = sC[1 * 16 + l15][kb]; b1.y = sC[1 * 16 + l15][kb + 1];
            v2f b2; b2.x = sC[2 * 16 + l15][kb]; b2.y = sC[2 * 16 + l15][kb + 1];
            v2f b3; b3.x = sC[3 * 16 + l15][kb]; b3.y = sC[3 * 16 + l15][kb + 1];
            acc0 = __builtin_amdgcn_wmma_f32_16x16x4_f32(false, a, false, b0, (short)0, acc0, false, false);
            acc1 = __builtin_amdgcn_wmma_f32_16x16x4_f32(false, a, false, b1, (short)0, acc1, false, false);
            acc2 = __builtin_amdgcn_wmma_f32_16x16x4_f32(false, a, false, b2, (short)0, acc2, false, false);
            acc3 = __builtin_amdgcn_wmma_f32_16x16x4_f32(false, a, false, b3, (short)0, acc3, false, false);
        }
        // logits = -d2 = -(f2 - 2*fc + c2)  (ALPHA = 1.0)
        v8f* accs[4] = { &acc0, &acc1, &acc2, &acc3 };
        #pragma unroll
        for (int nt = 0; nt < 4; ++nt) {
            #pragma unroll
            for (int v = 0; v < 8; ++v) {
                int mm = wave * 16 + v + (lhalf << 3);
                int nn = nt * 16 + l15;
                float fc = (*accs[nt])[v];
                sSim[mm][nn] = 2.0f * fc - sF2[mm] - sC2[nn];
            }
        }
    }
    __syncthreads();

    // ---- softmax over K per row ----
    if (tid < ROWS) {
        float mx = -3.402823466e38f;
        for (int k = 0; k < NK; ++k) mx = fmaxf(mx, sSim[tid][k]);
        float sum = 0.0f;
        for (int k = 0; k < NK; ++k) {
            float e = __expf(sSim[tid][k] - mx);
            sSim[tid][k] = e;
            sum += e;
        }
        float inv = 1.0f / sum;
        for (int k = 0; k < NK; ++k) sSim[tid][k] *= inv;
    }
    __syncthreads();

    // ---- S0 partial: sum of sim over rows ----
    if (tid < NK) {
        float s = 0.0f;
        for (int r = 0; r < ROWS; ++r) s += sSim[r][tid];
        atomicAdd(&S0[b * NK + tid], s);
    }

    // ---- GEMM 2: S1[k][d] += sum_r sim[r][k] * F[r][d]
    //      M=64 (4 tiles), N=128 (8 tiles) -> wave owns (mt = w&3, nt = 4*(w>>2)..+3)
    {
        const int mt  = wave & 3;
        const int ntb = (wave >> 2) * 4;
        v8f acc0 = {}, acc1 = {}, acc2 = {}, acc3 = {};
        const int kcol = mt * 16 + l15;
        for (int r0 = 0; r0 < ROWS; r0 += 4) {
            const int rr = r0 + (lhalf << 1);
            v2f a; a.x = sSim[rr][kcol]; a.y = sSim[rr + 1][kcol];  // A = sim^T
            v2f b0; b0.x = sF[rr][(ntb + 0) * 16 + l15]; b0.y = sF[rr + 1][(ntb + 0) * 16 + l15];
            v2f b1; b1.x = sF[rr][(ntb + 1) * 16 + l15]; b1.y = sF[rr + 1][(ntb + 1) * 16 + l15];
            v2f b2; b2.x = sF[rr][(ntb + 2) * 16 + l15]; b2.y = sF[rr + 1][(ntb + 2) * 16 + l15];
            v2f b3; b3.x = sF[rr][(ntb + 3) * 16 + l15]; b3.y = sF[rr + 1][(ntb + 3) * 16 + l15];
            acc0 = __builtin_amdgcn_wmma_f32_16x16x4_f32(false, a, false, b0, (short)0, acc0, false, false);
            acc1 = __builtin_amdgcn_wmma_f32_16x16x4_f32(false, a, false, b1, (short)0, acc1, false, false);
            acc2 = __builtin_amdgcn_wmma_f32_16x16x4_f32(false, a, false, b2, (short)0, acc2, false, false);
            acc3 = __builtin_amdgcn_wmma_f32_16x16x4_f32(false, a, false, b3, (short)0, acc3, false, false);
        }
        v8f* accs[4] = { &acc0, &acc1, &acc2, &acc3 };
        #pragma unroll
        for (int j = 0; j < 4; ++j) {
            #pragma unroll
            for (int v = 0; v < 8; ++v) {
                int k = mt * 16 + v + (lhalf << 3);
                int d = (ntb + j) * 16 + l15;
                atomicAdd(&S1[((size_t)b * NK + k) * ND + d], (*accs[j])[v]);
            }
        }
    }
}

// vlad = S1 - S0*c ; store pre-norm output ; accumulate global sum of squares
__global__ __launch_bounds__(256) void netvlad_vlad_reduce(
    const float* __restrict__ S1, const float* __restrict__ S0,
    const float* __restrict__ cent, float* __restrict__ out,
    float* __restrict__ nrm)
{
    __shared__ float red[256];
    int idx = blockIdx.x * 256 + threadIdx.x;
    int b = idx >> 13;             // / (K*D)
    int k = (idx >> 7) & (NK - 1); // / D % K
    float v = S1[idx] - S0[b * NK + k] * cent[idx];
    out[idx] = v;
    red[threadIdx.x] = v * v;
    __syncthreads();
    for (int s = 128; s > 0; s >>= 1) {
        if (threadIdx.x < s) red[threadIdx.x] += red[threadIdx.x + s];
        __syncthreads();
    }
    if (threadIdx.x == 0) atomicAdd(nrm, red[0]);
}

__global__ void netvlad_normalize(float* __restrict__ out, const float* __restrict__ nrm) {
    int idx = blockIdx.x * blockDim.x + threadIdx.x;
    float s = rsqrtf(fmaxf(nrm[0], 1e-12f));
    out[idx] *= s;
}

extern "C" void kernel_launch(void* const* d_in, const int* in_sizes, int n_in,
                              void* d_out, int out_size, void* d_ws, size_t ws_size,
                              hipStream_t stream) {
    const float* feat = (const float*)d_in[0];   // (32, 32, 32, 128)
    const float* cent = (const float*)d_in[1];   // (32, 64, 128)
    float* out = (float*)d_out;                  // (32, 8192)
    float* ws  = (float*)d_ws;

    float* S1  = ws;                 // B*K*D
    float* S0  = ws + S1_ELEMS;      // B*K
    float* nrm = S0 + S0_ELEMS;      // 1

    netvlad_zero_ws<<<(WS_ZERO_ELEMS + 255) / 256, 256, 0, stream>>>(ws, WS_ZERO_ELEMS);
    netvlad_sim_acc<<<NB * CHUNKS, 256, 0, stream>>>(feat, cent, S1, S0);
    netvlad_vlad_reduce<<<S1_ELEMS / 256, 256, 0, stream>>>(S1, S0, cent, out, nrm);
    netvlad_normalize<<<S1_ELEMS / 256, 256, 0, stream>>>(out, nrm);
}